// SparseFcos3DHead_61246233641489
// MI455X (gfx1250) — compile-verified
//
#include <hip/hip_runtime.h>
#include <hip/hip_bf16.h>
#include <stdint.h>

// ---------------------------------------------------------------------------
// fcaf3d-style head post-processing for MI455X (gfx1250).
//   K0 score_hist : streaming fused-score pass, cls tiles staged to LDS via
//                   global_load_async_to_lds_b128 (ASYNCcnt), 4096-bin hist
//   K1 thresh     : per-level score threshold from histogram suffix-sum
//   K2 compact    : gather candidate indices >= threshold
//   K3 sortdecode : per-level bitonic top-1000 + WMMA (v_wmma_f32_16x16x4_f32)
//                   box decode ( [points|pred](16x12) @ W(12x16) ) + score gather
//   K4 nms        : per-class bitonic top-500 + greedy 3D-IoU NMS + outputs
// ---------------------------------------------------------------------------

#define N_CLASSES 18
#define NMS_PRE   1000
#define K_NMS     500
#define SCORE_THR 0.01f
#define IOU_THR   0.5f
#define CAP       4096
#define NBINS     4096
#define TOT_CAND  (4 * NMS_PRE)   // 4000

typedef __attribute__((ext_vector_type(2))) float v2f;
typedef __attribute__((ext_vector_type(8))) float v8f;

__device__ __forceinline__ float sigmoidf_(float x) {
    return 1.0f / (1.0f + __expf(-x));
}

// ---------------------------------------------------------------------------
// K_init: zero histograms + counters
// ---------------------------------------------------------------------------
__global__ void fcaf_init_kernel(unsigned* hist, unsigned* cnt) {
    int i = blockIdx.x * blockDim.x + threadIdx.x;
    if (i < 4 * NBINS) hist[i] = 0u;
    if (i < 4) cnt[i] = 0u;
}

// ---------------------------------------------------------------------------
// K0: fused score + histogram. cls tile (256 x 18 f32 = 1152 x b128) staged
// into LDS with async global->LDS copies (CDNA5 ASYNCcnt path).
// ---------------------------------------------------------------------------
#define SB 256
__global__ void fcaf_score_hist_kernel(const float* __restrict__ cn,
                                       const float* __restrict__ cls,
                                       float* __restrict__ scores,
                                       unsigned* __restrict__ hist,
                                       int n) {
    __shared__ __align__(16) float tile[SB * 18];
    __shared__ unsigned lhist[NBINS];
    const int tid = threadIdx.x;
    for (int i = tid; i < NBINS; i += blockDim.x) lhist[i] = 0u;
    __syncthreads();

    for (int base = blockIdx.x * SB; base < n; base += gridDim.x * SB) {
        const int rem = n - base;
        float m = -3.0e38f;
        if (rem >= SB) {
            __syncthreads();  // previous tile fully consumed
            const float* gsrc = cls + (size_t)base * 18;  // 16B aligned (base%256==0)
            for (int c = tid; c < (SB * 18) / 4; c += SB) {
                unsigned loff = (unsigned)(uintptr_t)(&tile[c * 4]);
                unsigned goff = (unsigned)(c * 16);
                asm volatile("global_load_async_to_lds_b128 %0, %1, %2"
                             :: "v"(loff), "v"(goff), "s"(gsrc)
                             : "memory");
            }
            asm volatile("s_wait_asynccnt 0" ::: "memory");
            __syncthreads();
            m = tile[tid * 18];
#pragma unroll
            for (int c = 1; c < N_CLASSES; ++c) m = fmaxf(m, tile[tid * 18 + c]);
        } else {
            int p = base + tid;
            if (p < n) {
                const float* row = cls + (size_t)p * 18;
                m = row[0];
#pragma unroll
                for (int c = 1; c < N_CLASSES; ++c) m = fmaxf(m, row[c]);
            }
        }
        int p = base + tid;
        if (p < n) {
            float s = sigmoidf_(m) * sigmoidf_(cn[p]);
            scores[p] = s;
            int b = (int)(s * (float)NBINS);
            b = min(max(b, 0), NBINS - 1);
            atomicAdd(&lhist[b], 1u);
        }
    }
    __syncthreads();
    for (int i = tid; i < NBINS; i += blockDim.x)
        if (lhist[i]) atomicAdd(&hist[i], lhist[i]);
}

// ---------------------------------------------------------------------------
// K1: per-level threshold (one block per level)
// ---------------------------------------------------------------------------
__global__ void fcaf_thresh_kernel(const unsigned* __restrict__ hist,
                                   float* __restrict__ thr) {
    if (threadIdx.x == 0) {
        const unsigned* h = hist + blockIdx.x * NBINS;
        unsigned acc = 0;
        int b = NBINS - 1;
        for (; b >= 0; --b) {
            acc += h[b];
            if (acc >= NMS_PRE) break;
        }
        if (b < 0) b = 0;
        thr[blockIdx.x] = (float)b * (1.0f / (float)NBINS);
    }
}

// ---------------------------------------------------------------------------
// K2: compaction of candidate indices >= threshold
// ---------------------------------------------------------------------------
__global__ void fcaf_compact_kernel(const float* __restrict__ scores,
                                    const float* __restrict__ thr,
                                    unsigned* __restrict__ cnt,
                                    int* __restrict__ cand,
                                    int n, int level) {
    float t = thr[level];
    for (int i = blockIdx.x * blockDim.x + threadIdx.x; i < n;
         i += gridDim.x * blockDim.x) {
        if (scores[i] >= t) {
            unsigned p = atomicAdd(&cnt[level], 1u);
            if (p < CAP) cand[level * CAP + p] = i;
        }
    }
}

// ---------------------------------------------------------------------------
// bitonic sort (descending by key), n must be power of two, in LDS
// ---------------------------------------------------------------------------
__device__ void bitonic_desc(float* key, int* idx, int n) {
    for (int k = 2; k <= n; k <<= 1) {
        for (int j = k >> 1; j > 0; j >>= 1) {
            __syncthreads();
            for (int i = threadIdx.x; i < n; i += blockDim.x) {
                int ixj = i ^ j;
                if (ixj > i) {
                    float ki = key[i], kj = key[ixj];
                    bool sw = ((i & k) == 0) ? (ki < kj) : (ki > kj);
                    if (sw) {
                        key[i] = kj; key[ixj] = ki;
                        int ti = idx[i]; idx[i] = idx[ixj]; idx[ixj] = ti;
                    }
                }
            }
        }
    }
    __syncthreads();
}

// constant decode matrix W[k][n]:  bbox = [px py pz d0..d5] @ W
__device__ __forceinline__ float wconst(int k, int n) {
    if (k < 3) return (n == k) ? 1.0f : 0.0f;
    if (k > 8) return 0.0f;
    int d = k - 3, axis = d >> 1;
    if (n == axis) return (d & 1) ? 0.5f : -0.5f;       // center shift
    if (n == 3 + axis) return 1.0f;                      // size = d- + d+
    return 0.0f;
}

__device__ __forceinline__ float aval(const float* pts, const float* pred,
                                      int idx, int k) {
    if (idx < 0 || k > 8) return 0.0f;
    if (k < 3) return pts[(size_t)idx * 3 + k];
    return pred[(size_t)idx * 6 + (k - 3)];
}

// ---------------------------------------------------------------------------
// K3: one block per level. bitonic top-1000 then WMMA decode (16 boxes/tile,
// 3 chained v_wmma_f32_16x16x4_f32 per tile) + per-class score gather.
// ---------------------------------------------------------------------------
__global__ __launch_bounds__(1024)
void fcaf_sortdecode_kernel(const float* __restrict__ cn,
                            const float* __restrict__ cls,
                            const float* __restrict__ pred,
                            const float* __restrict__ pts,
                            const float* __restrict__ scores,
                            const int* __restrict__ cand,
                            const unsigned* __restrict__ cnt,
                            int level,
                            float* __restrict__ mbb,   // [4000,6]
                            float* __restrict__ msc) { // [4000,18]
    __shared__ float skey[CAP];
    __shared__ int sidx[CAP];
    const int tid = threadIdx.x;
    const int valid = min((int)cnt[level], CAP);

    for (int i = tid; i < CAP; i += blockDim.x) {
        if (i < valid) {
            int id = cand[level * CAP + i];
            skey[i] = scores[id];
            sidx[i] = id;
        } else {
            skey[i] = -3.0e38f;
            sidx[i] = -1;
        }
    }
    bitonic_desc(skey, sidx, CAP);

    // ---- WMMA decode: tiles of 16 boxes, K = 12 (9 real + 3 zero pad) ----
    const int wave = tid >> 5;
    const int lane = tid & 31;
    const int half = lane >> 4;
    const int nn = lane & 15;
    const int lim = min(valid, NMS_PRE);
    for (int t = wave; t < (NMS_PRE + 15) / 16; t += 32) {
        v8f acc = {0.f, 0.f, 0.f, 0.f, 0.f, 0.f, 0.f, 0.f};
        int row = t * 16 + nn;
        int idx = (row < lim) ? sidx[row] : -1;
#pragma unroll
        for (int kk = 0; kk < 3; ++kk) {
            int k0 = kk * 4 + half * 2;   // A: lanes0-15 hold K0/K1, lanes16-31 K2/K3
            v2f a, b;
            a.x = aval(pts, pred, idx, k0);
            a.y = aval(pts, pred, idx, k0 + 1);
            b.x = wconst(k0, nn);         // B: same K striping, N = lane&15
            b.y = wconst(k0 + 1, nn);
            acc = __builtin_amdgcn_wmma_f32_16x16x4_f32(
                false, a, false, b, (short)0, acc, false, false);
        }
        if (nn < 6) {
#pragma unroll
            for (int r = 0; r < 8; ++r) {
                int m = half * 8 + r;     // D: lanes0-15 rows 0-7, lanes16-31 rows 8-15
                int j = t * 16 + m;
                if (j < NMS_PRE)
                    mbb[((size_t)(level * NMS_PRE + j)) * 6 + nn] = acc[r];
            }
        }
    }
    __syncthreads();

    // ---- per-class fused score gather for the selected 1000 ----
    for (int q = tid; q < NMS_PRE * N_CLASSES; q += blockDim.x) {
        int j = q / N_CLASSES, c = q % N_CLASSES;
        int id = (j < lim) ? sidx[j] : -1;
        float s = 0.0f;
        if (id >= 0)
            s = sigmoidf_(cls[(size_t)id * N_CLASSES + c]) * sigmoidf_(cn[id]);
        msc[(size_t)(level * NMS_PRE + j) * N_CLASSES + c] = s;
    }
}

// ---------------------------------------------------------------------------
// K4: per-class (18 blocks) top-500 + greedy axis-aligned 3D NMS
// ---------------------------------------------------------------------------
__device__ __forceinline__ float iou3d(const float* a, const float* b) {
    float inter = 1.0f, va = 1.0f, vb = 1.0f;
#pragma unroll
    for (int d = 0; d < 3; ++d) {
        float amn = a[d] - a[3 + d] * 0.5f, amx = a[d] + a[3 + d] * 0.5f;
        float bmn = b[d] - b[3 + d] * 0.5f, bmx = b[d] + b[3 + d] * 0.5f;
        inter *= fmaxf(fminf(amx, bmx) - fmaxf(amn, bmn), 0.0f);
        va *= fmaxf(amx - amn, 0.0f);
        vb *= fmaxf(bmx - bmn, 0.0f);
    }
    return inter / (va + vb - inter + 1e-6f);
}

__global__ __launch_bounds__(1024)
void fcaf_nms_kernel(const float* __restrict__ mbb,   // [4000,6]
                     const float* __restrict__ msc,   // [4000,18]
                     float* __restrict__ out) {
    const int c = blockIdx.x;     // class
    const int tid = threadIdx.x;
    __shared__ float skey[CAP];
    __shared__ int sidx[CAP];
    __shared__ float bx[K_NMS * 6];
    __shared__ int sup[K_NMS];
    __shared__ int vld[K_NMS];

    for (int i = tid; i < CAP; i += blockDim.x) {
        if (i < TOT_CAND) {
            float s = msc[(size_t)i * N_CLASSES + c];
            skey[i] = (s > SCORE_THR) ? s : -1.0f;   // matches where(valid, s, -1)
            sidx[i] = i;
        } else {
            skey[i] = -2.0f;
            sidx[i] = 0;
        }
    }
    bitonic_desc(skey, sidx, CAP);

    for (int j = tid; j < K_NMS; j += blockDim.x) {
        int id = sidx[j];
#pragma unroll
        for (int q = 0; q < 6; ++q) bx[j * 6 + q] = mbb[(size_t)id * 6 + q];
        int v = skey[j] > SCORE_THR;
        vld[j] = v;
        sup[j] = !v;
    }

    for (int i = 0; i < K_NMS; ++i) {
        __syncthreads();
        if (!sup[i]) {  // block-uniform branch
            for (int j = tid; j < K_NMS; j += blockDim.x)
                if (j > i && iou3d(&bx[i * 6], &bx[j * 6]) > IOU_THR) sup[j] = 1;
        }
    }
    __syncthreads();

    // outputs: boxes [18*500*6] | scores [9000] | labels [9000] | keep [9000]
    float* ob = out;
    float* os = out + N_CLASSES * K_NMS * 6;
    float* ol = os + N_CLASSES * K_NMS;
    float* ok = ol + N_CLASSES * K_NMS;
    for (int j = tid; j < K_NMS; j += blockDim.x) {
        int g = c * K_NMS + j;
#pragma unroll
        for (int q = 0; q < 6; ++q) ob[(size_t)g * 6 + q] = bx[j * 6 + q];
        os[g] = skey[j];
        ol[g] = (float)c;
        ok[g] = (!sup[j] && vld[j]) ? 1.0f : 0.0f;
    }
}

// ---------------------------------------------------------------------------
extern "C" void kernel_launch(void* const* d_in, const int* in_sizes, int n_in,
                              void* d_out, int out_size, void* d_ws, size_t ws_size,
                              hipStream_t stream) {
    (void)n_in; (void)out_size; (void)ws_size;
    const float* cn[4];  const float* bp[4];
    const float* cs[4];  const float* pt[4];
    int n[4]; long off[4]; long tot = 0;
    for (int l = 0; l < 4; ++l) {
        cn[l] = (const float*)d_in[4 * l + 0];
        bp[l] = (const float*)d_in[4 * l + 1];
        cs[l] = (const float*)d_in[4 * l + 2];
        pt[l] = (const float*)d_in[4 * l + 3];
        n[l] = in_sizes[4 * l];      // centerness is [n,1]
        off[l] = tot;
        tot += n[l];
    }

    // workspace layout (all 4-byte aligned)
    float*    ws_scores = (float*)d_ws;                 // tot floats
    unsigned* hist      = (unsigned*)(ws_scores + tot); // 4*NBINS
    unsigned* cnt       = hist + 4 * NBINS;             // 4
    float*    thr       = (float*)(cnt + 4);            // 4
    int*      cand      = (int*)(thr + 4);              // 4*CAP
    float*    mbb       = (float*)(cand + 4 * CAP);     // 4000*6
    float*    msc       = mbb + TOT_CAND * 6;           // 4000*18

    fcaf_init_kernel<<<(4 * NBINS + 255) / 256, 256, 0, stream>>>(hist, cnt);

    for (int l = 0; l < 4; ++l) {
        int blocks = min((n[l] + SB - 1) / SB, 768);
        fcaf_score_hist_kernel<<<blocks, SB, 0, stream>>>(
            cn[l], cs[l], ws_scores + off[l], hist + l * NBINS, n[l]);
    }

    fcaf_thresh_kernel<<<4, 64, 0, stream>>>(hist, thr);

    for (int l = 0; l < 4; ++l) {
        int blocks = min((n[l] + 255) / 256, 768);
        fcaf_compact_kernel<<<blocks, 256, 0, stream>>>(
            ws_scores + off[l], thr, cnt, cand, n[l], l);
    }

    for (int l = 0; l < 4; ++l) {
        fcaf_sortdecode_kernel<<<1, 1024, 0, stream>>>(
            cn[l], cs[l], bp[l], pt[l], ws_scores + off[l], cand, cnt, l,
            mbb, msc);
    }

    fcaf_nms_kernel<<<N_CLASSES, 1024, 0, stream>>>(mbb, msc, (float*)d_out);
}